// PatchTransformer_74912819577619
// MI455X (gfx1250) — compile-verified
//
#include <hip/hip_runtime.h>
#include <stdint.h>

// ---------------- problem constants (from reference) ----------------
#define PATCHN 300
#define BB     16
#define FF     14
#define NBF    (BB * FF)          // 224
#define KER    7
#define PADW   3
#define CH     3
#define CHW    (PATCHN * PATCHN)  // 90000
#define INNER  (CH * CHW)         // 270000 elements per (b,f)
#define TILE   16
#define LTILE  (TILE + 2 * PADW)  // 22

typedef float v4f __attribute__((ext_vector_type(4)));

// CDNA5 async global->LDS path (ASYNCcnt), guarded so build never breaks.
// Probe round 1 revealed the builtin's param 0 type: '__device__ int *'
// (i.e. int in address space 1, non-const).
#if defined(__HIP_DEVICE_COMPILE__) && __has_builtin(__builtin_amdgcn_global_load_async_to_lds_b32)
#define HAVE_ASYNC_LDS 1
typedef __attribute__((address_space(1))) int as1_int;
typedef __attribute__((address_space(3))) int as3_int;
#endif

__device__ __forceinline__ int reflect_idx(int i, int n) {
  // jnp.pad 'reflect': mirror WITHOUT repeating the edge sample
  if (i < 0) i = -i;
  if (i >= n) i = 2 * n - 2 - i;
  return i;
}

__device__ __forceinline__ void cswap(float& a, float& b) {
  float lo = fminf(a, b);
  float hi = fmaxf(a, b);
  a = lo;
  b = hi;
}

// ---------------- Kernel A: 7x7 median with reflect padding ----------------
// One 16x16 block per output tile per channel. Halo tile staged in LDS via
// CDNA5 async load-to-LDS when available.
__global__ __launch_bounds__(TILE* TILE) void median7_kernel(
    const float* __restrict__ patch, float* __restrict__ med) {
  __shared__ float tile[LTILE * LTILE];

  const int tx  = threadIdx.x;
  const int ty  = threadIdx.y;
  const int tid = ty * TILE + tx;
  const int x0  = blockIdx.x * TILE;
  const int y0  = blockIdx.y * TILE;
  const int c   = blockIdx.z;

  const float* __restrict__ src = patch + c * CHW;

  // Fill 22x22 halo tile (484 elements) with 256 threads.
  for (int k = tid; k < LTILE * LTILE; k += TILE * TILE) {
    const int ly = k / LTILE;
    const int lx = k - ly * LTILE;
    const int gy = reflect_idx(y0 + ly - PADW, PATCHN);
    const int gx = reflect_idx(x0 + lx - PADW, PATCHN);
    const float* g = src + gy * PATCHN + gx;
#ifdef HAVE_ASYNC_LDS
    __builtin_amdgcn_global_load_async_to_lds_b32((as1_int*)g,
                                                  (as3_int*)&tile[k],
                                                  /*offset=*/0, /*cpol=*/0);
#else
    tile[k] = *g;
#endif
  }
#ifdef HAVE_ASYNC_LDS
#if __has_builtin(__builtin_amdgcn_s_wait_asynccnt)
  __builtin_amdgcn_s_wait_asynccnt(0);
#else
  asm volatile("s_wait_asynccnt 0" ::: "memory");
#endif
#endif
  __syncthreads();

  // Gather 7x7 window into registers (static indices -> no scratch).
  float v[KER * KER];
#pragma unroll
  for (int dy = 0; dy < KER; ++dy) {
#pragma unroll
    for (int dx = 0; dx < KER; ++dx) {
      v[dy * KER + dx] = tile[(ty + dy) * LTILE + (tx + dx)];
    }
  }

  // Odd-even transposition sort, fully unrolled: 49 passes guarantee sorted.
#pragma unroll
  for (int p = 0; p < KER * KER; ++p) {
#pragma unroll
    for (int i = (p & 1); i + 1 < KER * KER; i += 2) {
      cswap(v[i], v[i + 1]);
    }
  }

  const int ox = x0 + tx;
  const int oy = y0 + ty;
  if (ox < PATCHN && oy < PATCHN) {
    med[c * CHW + oy * PATCHN + ox] = v[24];  // MED_IDX = 24
  }
}

// ---------------- Kernel B: broadcast * contrast + brightness + 0.1*noise, clamp ----
// Pure bandwidth: float4 streams, NT hints on single-use noise/output, regular
// (temporal) loads for the 1 MB median tile so it stays hot in L2 across 224 reuses.
__global__ __launch_bounds__(256) void transform_kernel(
    const float* __restrict__ med, const float* __restrict__ contrast,
    const float* __restrict__ brightness, const float* __restrict__ noise,
    float* __restrict__ out) {
  const int bf = blockIdx.y;                               // (b*F + f), 0..223
  const int i  = blockIdx.x * blockDim.x + threadIdx.x;    // float4 index
  if (i >= INNER / 4) return;

  const float con = contrast[bf];
  const float br  = brightness[bf];

  const int base = bf * INNER + i * 4;                     // < 2^31, 16B aligned

  v4f nz = __builtin_nontemporal_load((const v4f*)(noise + base));
  v4f md = *(const v4f*)(med + i * 4);

  v4f r;
#pragma unroll
  for (int k = 0; k < 4; ++k) {
    float t = fmaf(md[k], con, fmaf(nz[k], 0.1f, br));
    r[k]    = fminf(fmaxf(t, 1e-6f), 0.99999f);
  }
  __builtin_nontemporal_store(r, (v4f*)(out + base));
}

// ---------------- launch ----------------
extern "C" void kernel_launch(void* const* d_in, const int* in_sizes, int n_in,
                              void* d_out, int out_size, void* d_ws, size_t ws_size,
                              hipStream_t stream) {
  const float* adv_patch  = (const float*)d_in[0];
  // d_in[1] = lab_batch : shape-only in reference, unused
  const float* contrast   = (const float*)d_in[2];
  const float* brightness = (const float*)d_in[3];
  const float* noise      = (const float*)d_in[4];
  // d_in[5] = img_size : unused

  float* med = (float*)d_ws;   // 270000 floats = 1.08 MB scratch
  float* out = (float*)d_out;

  dim3 blockA(TILE, TILE, 1);
  dim3 gridA((PATCHN + TILE - 1) / TILE, (PATCHN + TILE - 1) / TILE, CH);
  median7_kernel<<<gridA, blockA, 0, stream>>>(adv_patch, med);

  dim3 gridB((INNER / 4 + 255) / 256, NBF, 1);  // (264, 224)
  transform_kernel<<<gridB, 256, 0, stream>>>(med, contrast, brightness, noise, out);
}